// RmiNet_22419729286022
// MI455X (gfx1250) — compile-verified
//
#include <hip/hip_runtime.h>
#include <math.h>

typedef __attribute__((ext_vector_type(2))) float v2f;
typedef __attribute__((ext_vector_type(8))) float v8f;

#define WAVES   4
#define ROWS    16
#define HPITCH  197   // conv buffer pitch  (mod 64 = 5,  bank-conflict-free)
#define H1PITCH 67    // h1 pitch           (mod 64 = 3,  bank-conflict-free)
#define H2PITCH 33    // h2 pitch           (mod 64 = 33, odd -> conflict-free)
#define OPITCH  17
#define H2OFF   1536  // h1 uses [0,1072); h2 at [1536,2064) within 3152-float wave slab

// One cyclic-Jacobi rotation on symmetric S (3x3, row-major) accumulating V.
// Only called with constant (p,q): after inlining all indices are constants,
// so the arrays are SROA'd into VGPRs.
__device__ __forceinline__ void jacobi3(float* s, float* v, int p, int q) {
  float apq = s[p * 3 + q];
  if (fabsf(apq) < 1e-20f) return;
  float app = s[p * 3 + p], aqq = s[q * 3 + q];
  float tau = (aqq - app) / (2.f * apq);
  float t  = copysignf(1.f, tau) / (fabsf(tau) + sqrtf(1.f + tau * tau));
  float c  = 1.f / sqrtf(1.f + t * t);
  float sn = t * c;
  const int k = 3 - p - q;
  float skp = s[k * 3 + p], skq = s[k * 3 + q];
  float nkp = c * skp - sn * skq;
  float nkq = sn * skp + c * skq;
  s[k * 3 + p] = nkp; s[p * 3 + k] = nkp;
  s[k * 3 + q] = nkq; s[q * 3 + k] = nkq;
  s[p * 3 + p] = c * c * app - 2.f * sn * c * apq + sn * sn * aqq;
  s[q * 3 + q] = sn * sn * app + 2.f * sn * c * apq + c * c * aqq;
  s[p * 3 + q] = 0.f; s[q * 3 + p] = 0.f;
#pragma unroll
  for (int i = 0; i < 3; ++i) {
    float vip = v[i * 3 + p], viq = v[i * 3 + q];
    v[i * 3 + p] = c * vip - sn * viq;
    v[i * 3 + q] = sn * vip + c * viq;
  }
}

__global__ __launch_bounds__(128) void rmi_fused(
    const float* __restrict__ x,   const float* __restrict__ cwg,
    const float* __restrict__ cbg, const float* __restrict__ w1,
    const float* __restrict__ b1,  const float* __restrict__ w2,
    const float* __restrict__ b2,  const float* __restrict__ w3,
    const float* __restrict__ b3,  float* __restrict__ out, int B) {
  __shared__ float lds [WAVES * ROWS * HPITCH];   // conv buf, later h1/h2
  __shared__ float ldsO[WAVES * ROWS * OPITCH];   // final 16x16 out tile

  const int tid  = threadIdx.x;
  const int wave = tid >> 5;
  const int lane = tid & 31;
  const int rowBase = blockIdx.x * (WAVES * ROWS) + wave * ROWS;
  float* H = &lds [wave * ROWS * HPITCH];
  float* O = &ldsO[wave * ROWS * OPITCH];

  // conv weights -> registers (tiny, L2-resident)
  float cw[30];
#pragma unroll
  for (int i = 0; i < 30; ++i) cw[i] = cwg[i];
  const float cb = cbg[0];

  // ---- phase 1: conv1d 'valid' [16 rows][196] into LDS -------------------
  for (int r = 0; r < ROWS; ++r) {
    int sr = rowBase + r; sr = (sr < B) ? sr : (B - 1);
    const float* xr = x + (size_t)sr * 1200;
    for (int j = lane; j < 196; j += 32) {
      __builtin_prefetch(xr + j + 256, 0, 1);   // global_prefetch_b8
      float acc = cb;
#pragma unroll
      for (int c = 0; c < 6; ++c)
#pragma unroll
        for (int k = 0; k < 5; ++k)
          acc = fmaf(xr[c * 200 + j + k], cw[c * 5 + k], acc);
      H[r * HPITCH + j] = acc;
    }
  }
  __syncthreads();

  // WMMA fragment coordinates (f32 16x16x4 layouts, ISA 7.12.2)
  const int hs = (lane >= 16) ? 2 : 0;   // K half-select for A/B fragments
  const int rA = lane & 15;              // M (A) / N (B,C) index
  const int rD = (lane < 16) ? 0 : 8;    // C/D row offset

  // Clamped column indices: padded B columns may read duplicate (finite) data;
  // the padded D columns are either overwritten with zero or never consumed.
  int cc1[4];
#pragma unroll
  for (int nt = 0; nt < 4; ++nt) {
    const int c = nt * 16 + rA;
    cc1[nt] = (c < 50) ? c : 49;
  }

  // ---- phase 2: GEMM1 [16,196] x [196,50(pad 64)], branch-free body ------
  v8f acc1[4] = {};
#pragma unroll 2
  for (int kk = 0; kk < 49; ++kk) {
    const int k0 = kk * 4 + hs;                       // 0..194, k0+1 <= 195
    v2f a; a.x = H[rA * HPITCH + k0]; a.y = H[rA * HPITCH + k0 + 1];
#pragma unroll
    for (int nt = 0; nt < 4; ++nt) {
      v2f b;
      b.x = w1[k0 * 50 + cc1[nt]];
      b.y = w1[(k0 + 1) * 50 + cc1[nt]];
      acc1[nt] = __builtin_amdgcn_wmma_f32_16x16x4_f32(
          false, a, false, b, (short)0, acc1[nt], false, false);
    }
  }
  __syncthreads();

  // bias+relu -> h1 [16][64] (cols >=50 forced to zero), reusing H region
  float* H1 = H;
#pragma unroll
  for (int nt = 0; nt < 4; ++nt) {
    const int col = nt * 16 + rA;
    const float bb = b1[cc1[nt]];
#pragma unroll
    for (int r = 0; r < 8; ++r) {
      float vv = fmaxf(acc1[nt][r] + bb, 0.f);
      vv = (col < 50) ? vv : 0.f;                     // zero the K padding
      H1[(rD + r) * H1PITCH + col] = vv;
    }
  }
  __syncthreads();

  // ---- phase 3: GEMM2 [16,52] x [50,30(pad 32)] --------------------------
  const int cc2 = (16 + rA < 30) ? (16 + rA) : 29;    // tile1 clamped col
  v8f acc2[2] = {};
  for (int kk = 0; kk < 13; ++kk) {
    const int k0  = kk * 4 + hs;                      // up to 51
    const int kc0 = (k0 < 50) ? k0 : 49;              // A[.,>=50]==0 anyway
    const int kc1 = (k0 + 1 < 50) ? (k0 + 1) : 49;
    v2f a;  a.x  = H1[rA * H1PITCH + k0]; a.y = H1[rA * H1PITCH + k0 + 1];
    v2f b0; b0.x = w2[kc0 * 30 + rA];     b0.y = w2[kc1 * 30 + rA];
    v2f b1v; b1v.x = w2[kc0 * 30 + cc2];  b1v.y = w2[kc1 * 30 + cc2];
    acc2[0] = __builtin_amdgcn_wmma_f32_16x16x4_f32(
        false, a, false, b0, (short)0, acc2[0], false, false);
    acc2[1] = __builtin_amdgcn_wmma_f32_16x16x4_f32(
        false, a, false, b1v, (short)0, acc2[1], false, false);
  }
  __syncthreads();
  float* H2 = H + H2OFF;
#pragma unroll
  for (int nt = 0; nt < 2; ++nt) {
    const int col = nt * 16 + rA;
    const float bb = b2[(col < 30) ? col : 29];
#pragma unroll
    for (int r = 0; r < 8; ++r) {
      float vv = fmaxf(acc2[nt][r] + bb, 0.f);
      vv = (col < 30) ? vv : 0.f;                     // zero the K padding
      H2[(rD + r) * H2PITCH + col] = vv;
    }
  }
  __syncthreads();

  // ---- phase 4: GEMM3 [16,32] x [30,15(pad 16)] --------------------------
  const int cc3 = (rA < 15) ? rA : 14;
  v8f acc3 = {};
  for (int kk = 0; kk < 8; ++kk) {
    const int k0  = kk * 4 + hs;                      // up to 31
    const int kc0 = (k0 < 30) ? k0 : 29;              // A[.,>=30]==0 anyway
    const int kc1 = (k0 + 1 < 30) ? (k0 + 1) : 29;
    v2f a; a.x = H2[rA * H2PITCH + k0]; a.y = H2[rA * H2PITCH + k0 + 1];
    v2f b; b.x = w3[kc0 * 15 + cc3];    b.y = w3[kc1 * 15 + cc3];
    acc3 = __builtin_amdgcn_wmma_f32_16x16x4_f32(
        false, a, false, b, (short)0, acc3, false, false);
  }
  {
    const float bb = b3[cc3];
#pragma unroll
    for (int r = 0; r < 8; ++r)                       // col 15 garbage: never read
      O[(rD + r) * OPITCH + rA] = acc3[r] + bb;
  }
  __syncthreads();

  // ---- phase 5: per-row Procrustes (3x3 Jacobi SVD), lanes 0..15 ---------
  if (lane < 16) {
    const int gRow = rowBase + lane;
    if (gRow < B) {
      float a9[9];
#pragma unroll
      for (int i = 0; i < 9; ++i) a9[i] = O[lane * OPITCH + i];
      // S = A^T A (symmetric), V = I
      float s[9], v[9];
#pragma unroll
      for (int i = 0; i < 3; ++i)
#pragma unroll
        for (int j = 0; j < 3; ++j) {
          s[i * 3 + j] = a9[0 * 3 + i] * a9[0 * 3 + j] +
                         a9[1 * 3 + i] * a9[1 * 3 + j] +
                         a9[2 * 3 + i] * a9[2 * 3 + j];
          v[i * 3 + j] = (i == j) ? 1.f : 0.f;
        }
#pragma unroll
      for (int sw = 0; sw < 6; ++sw) {
        jacobi3(s, v, 0, 1); jacobi3(s, v, 0, 2); jacobi3(s, v, 1, 2);
      }
      // sort sigma^2 descending, swapping V columns (constant indices only)
      float g0 = s[0], g1 = s[4], g2 = s[8];
#define SWAPCOL(p, q) { float t_;                              \
        t_ = v[p];     v[p]     = v[q];     v[q]     = t_;     \
        t_ = v[3 + p]; v[3 + p] = v[3 + q]; v[3 + q] = t_;     \
        t_ = v[6 + p]; v[6 + p] = v[6 + q]; v[6 + q] = t_; }
      if (g0 < g1) { float t = g0; g0 = g1; g1 = t; SWAPCOL(0, 1); }
      if (g0 < g2) { float t = g0; g0 = g2; g2 = t; SWAPCOL(0, 2); }
      if (g1 < g2) { float t = g1; g1 = g2; g2 = t; SWAPCOL(1, 2); }
#undef SWAPCOL
      // W = A * V ; Gram-Schmidt U from first two columns
      float w[9];
#pragma unroll
      for (int i = 0; i < 3; ++i)
#pragma unroll
        for (int j = 0; j < 3; ++j)
          w[i * 3 + j] = a9[i * 3 + 0] * v[0 + j] +
                         a9[i * 3 + 1] * v[3 + j] +
                         a9[i * 3 + 2] * v[6 + j];
      float u0x = w[0], u0y = w[3], u0z = w[6];
      float n0 = 1.f / sqrtf(u0x * u0x + u0y * u0y + u0z * u0z + 1e-20f);
      u0x *= n0; u0y *= n0; u0z *= n0;
      float u1x = w[1], u1y = w[4], u1z = w[7];
      float d01 = u1x * u0x + u1y * u0y + u1z * u0z;
      u1x -= d01 * u0x; u1y -= d01 * u0y; u1z -= d01 * u0z;
      float n1 = 1.f / sqrtf(u1x * u1x + u1y * u1y + u1z * u1z + 1e-20f);
      u1x *= n1; u1y *= n1; u1z *= n1;
      float cx = u0y * u1z - u0z * u1y;
      float cy = u0z * u1x - u0x * u1z;
      float cz = u0x * u1y - u0y * u1x;
      float detV = v[0] * (v[4] * v[8] - v[5] * v[7]) -
                   v[1] * (v[3] * v[8] - v[5] * v[6]) +
                   v[2] * (v[3] * v[7] - v[4] * v[6]);
      float sg = (detV >= 0.f) ? 1.f : -1.f;   // det(U_svd)*det(V) sign fix
      float u2x = sg * cx, u2y = sg * cy, u2z = sg * cz;
      // R = U * V^T
      float U_[9] = { u0x, u1x, u2x,  u0y, u1y, u2y,  u0z, u1z, u2z };
      float* op = out + (size_t)gRow * 15;
#pragma unroll
      for (int i = 0; i < 3; ++i)
#pragma unroll
        for (int j = 0; j < 3; ++j)
          op[i * 3 + j] = U_[i * 3 + 0] * v[j * 3 + 0] +
                          U_[i * 3 + 1] * v[j * 3 + 1] +
                          U_[i * 3 + 2] * v[j * 3 + 2];
#pragma unroll
      for (int i = 9; i < 15; ++i) op[i] = O[lane * OPITCH + i];
    }
  }
}

extern "C" void kernel_launch(void* const* d_in, const int* in_sizes, int n_in,
                              void* d_out, int out_size, void* d_ws, size_t ws_size,
                              hipStream_t stream) {
  (void)n_in; (void)out_size; (void)d_ws; (void)ws_size;
  const float* x   = (const float*)d_in[0];
  const float* cw  = (const float*)d_in[1];
  const float* cb  = (const float*)d_in[2];
  const float* w1  = (const float*)d_in[3];
  const float* b1  = (const float*)d_in[4];
  const float* w2  = (const float*)d_in[5];
  const float* b2  = (const float*)d_in[6];
  const float* w3  = (const float*)d_in[7];
  const float* b3  = (const float*)d_in[8];
  const int B = in_sizes[0] / 1200;               // x is [B,6,200]
  const int grid = (B + WAVES * ROWS - 1) / (WAVES * ROWS);
  rmi_fused<<<grid, WAVES * 32, 0, stream>>>(x, cw, cb, w1, b1, w2, b2, w3, b3,
                                             (float*)d_out, B);
}